// DetBlock_84447646974497
// MI455X (gfx1250) — compile-verified
//
#include <hip/hip_runtime.h>
#include <stdint.h>

#define GG        64
#define NA        3
#define NCLS      21
#define NCH       26                 // NCLS + 5
#define NTGT      2048
#define NB        64
#define CELLS     (GG * GG)          // 4096 cells per (b, anchor) slab
#define NSLAB     (NB * NA)          // 192
#define CHUNK     256
#define NCHUNK    (CELLS / CHUNK)    // 16
#define NBLK_MAIN (NSLAB * NCHUNK)   // 3072
#define NCELL     (NSLAB * CELLS)    // 786432
#define IGNORE_T  0.7f

typedef unsigned int v4u __attribute__((ext_vector_type(4)));
typedef int          v8i __attribute__((ext_vector_type(8)));
typedef int          v4i __attribute__((ext_vector_type(4)));

#if defined(__has_builtin)
#if __has_builtin(__builtin_amdgcn_tensor_load_to_lds) && __has_builtin(__builtin_amdgcn_s_wait_tensorcnt)
#define USE_TDM 1
#endif
#if __has_builtin(__builtin_amdgcn_tensor_store_from_lds)
#define USE_TDM_STORE 1
#endif
#endif
#ifndef USE_TDM
#define USE_TDM 0
#endif
#ifndef USE_TDM_STORE
#define USE_TDM_STORE 0
#endif

__device__ __forceinline__ float sigmoidf_fast(float x) {
    return 1.0f / (1.0f + __expf(-x));
}

__device__ __forceinline__ float bce(float p, float t) {
    p = fminf(fmaxf(p, 1e-7f), 1.0f - 1e-7f);
    return -(t * __logf(p) + (1.0f - t) * log1pf(-p));
}

__device__ __forceinline__ int cell_of(int b, int a, int gj, int gi) {
    return ((b * NA + a) * GG + gj) * GG + gi;
}

// Build TDM descriptor group0: {flags, lds_addr, global_addr lo, global_addr hi | type}
__device__ __forceinline__ v4u tdm_g0(uint32_t lds, uint64_t ga) {
    v4u g0;
    g0.x = 1u;                                                   // count = 1 valid descriptor
    g0.y = lds;                                                  // lds_addr (bytes)
    g0.z = (uint32_t)ga;                                         // global_addr[31:0]
    g0.w = (uint32_t)((ga >> 32) & 0x1FFFFFFull) | 0x80000000u;  // addr[56:32] | type=2
    return g0;
}

// Build TDM descriptor group1 for a 2-D f32 tile: tile_d0 x tile_d1, row stride s0 (elems)
__device__ __forceinline__ v8i tdm_g1(int tile_d0, int tile_d1, int s0) {
    v8i g1;
    g1[0] = 0x00020000;                 // data_size = 2 (4 bytes); no multicast/pad/iterate
    g1[1] = 0;                          // barrier addr = 0; tensor_dim0[15:0] = 0
    g1[2] = 0x4000;                     // tensor_dim0 = 1<<30 (no OOB clip); dim1 lo16 = 0
    g1[3] = (tile_d0 << 16) | 0x10;     // tile_dim0; tensor_dim1 = 1<<20
    g1[4] = tile_d1;                    // tile_dim1; tile_dim2 = 0 (2-D)
    g1[5] = s0;                         // tensor_dim0_stride (elements)
    g1[6] = 0;
    g1[7] = 0;
    return g1;
}

// ---------------------------------------------------------------- K0: init
__global__ __launch_bounds__(256) void k_init(int* __restrict__ twin,
                                              float* __restrict__ noobj) {
    int i = blockIdx.x * blockDim.x + threadIdx.x;
    if (i < NCELL) {
        twin[i]  = -1;
        noobj[i] = 1.0f;
    }
}

// ---------------------------------------------------------------- K1: targets
__global__ __launch_bounds__(256) void k_targets(const float* __restrict__ tg,
                                                 int* __restrict__ twin,
                                                 float* __restrict__ noobj,
                                                 float* __restrict__ tinfo) {
    int t = blockIdx.x * blockDim.x + threadIdx.x;
    if (t >= NTGT) return;
    const float* r = tg + t * 6;
    int   b   = (int)r[0];
    int   lbl = (int)r[1];
    float gxf = r[2] * (float)GG;
    float gyf = r[3] * (float)GG;
    float gw  = r[4] * (float)GG;
    float gh  = r[5] * (float)GG;
    int gi = (int)gxf;
    int gj = (int)gyf;

    // anchors in grid units (ANCHORS / stride, stride = 512/64 = 8)
    const float aw[NA] = {1.5f, 1.75f, 1.875f};
    const float ah[NA] = {3.25f, 3.625f, 4.25f};

    float iou[NA];
#pragma unroll
    for (int a = 0; a < NA; ++a) {
        float inter = fminf(aw[a], gw) * fminf(ah[a], gh);
        float uni   = aw[a] * ah[a] + gw * gh - inter;
        iou[a] = inter / uni;
    }
    int best = 0;                                // argmax, first-max wins
    if (iou[1] > iou[0]) best = 1;
    if (iou[2] > iou[best]) best = 2;

#pragma unroll
    for (int a = 0; a < NA; ++a)
        if (iou[a] > IGNORE_T) noobj[cell_of(b, a, gj, gi)] = 0.0f;

    int c = cell_of(b, best, gj, gi);
    noobj[c] = 0.0f;
    atomicMax(&twin[c], t);                      // deterministic scatter winner

    float* ti = tinfo + t * 8;
    ti[0] = gxf - floorf(gxf);
    ti[1] = gyf - floorf(gyf);
    ti[2] = __logf(gw / aw[best] + 1e-16f);
    ti[3] = __logf(gh / ah[best] + 1e-16f);
    ti[4] = (float)lbl;
}

// ---------------------------------------------------------------- K2: main stream
__global__ __launch_bounds__(256) void k_main(const float* __restrict__ x,
                                              float* __restrict__ out,
                                              const int* __restrict__ twin,
                                              const float* __restrict__ noobj,
                                              const float* __restrict__ tinfo,
                                              float* __restrict__ partials) {
    __shared__ float tile[NCH * CHUNK];   // 26 channel rows x 256 cells
    __shared__ float ostg[NCH * CHUNK];   // output staging, [cell][ch]
    __shared__ float red[8][6];

    const int tid   = threadIdx.x;
    const int blk   = blockIdx.x;
    const int slab  = blk >> 4;           // / NCHUNK
    const int chunk = blk & (NCHUNK - 1);
    const int b     = slab / NA;
    const int a     = slab - b * NA;
    const int col0  = chunk * CHUNK;

    // slab base element: x[b][a*26 + c][pos], channel row stride = 4096 floats
    const uint64_t slab_elem = ((uint64_t)b * (NA * NCH) + (uint64_t)a * NCH) * CELLS + (uint64_t)col0;

#if USE_TDM
    {
        // ---- TDM load: 26 rows x 256 f32 tile, row stride 4096 f32
        uint64_t ga  = (uint64_t)(const void*)x + slab_elem * 4ull;
        uint32_t lds = (uint32_t)(uint64_t)(void*)&tile[0];   // low 32 bits = LDS offset
        v4u g0 = tdm_g0(lds, ga);
        v8i g1 = tdm_g1(CHUNK, NCH, CELLS);
        v4i gz4 = {0, 0, 0, 0};
        v8i gz8 = {0, 0, 0, 0, 0, 0, 0, 0};
        if (tid == 0)
            __builtin_amdgcn_tensor_load_to_lds(g0, g1, gz4, gz4, gz8, 0);
    }
    __builtin_amdgcn_s_wait_tensorcnt(0);
    __syncthreads();
#else
    for (int i = tid; i < NCH * CHUNK; i += 256) {
        int c   = i >> 8;          // / CHUNK
        int col = i & (CHUNK - 1);
        tile[i] = x[slab_elem + (uint64_t)c * CELLS + col];
    }
    __syncthreads();
#endif

    // ---- per-cell math (one thread = one cell)
    float v[NCH];
#pragma unroll
    for (int c = 0; c < NCH; ++c) v[c] = tile[c * CHUNK + tid];

    const int   pos = col0 + tid;
    const float gx  = (float)(pos & (GG - 1));
    const float gy  = (float)(pos >> 6);

    const float awpx[NA] = {12.0f, 14.0f, 15.0f};   // anchors in pixels
    const float ahpx[NA] = {26.0f, 29.0f, 34.0f};

    const float s0 = sigmoidf_fast(v[0]);
    const float s1 = sigmoidf_fast(v[1]);
    float sc[NCLS + 1];
#pragma unroll
    for (int c = 0; c < NCLS + 1; ++c) sc[c] = sigmoidf_fast(v[4 + c]);

    // output staging: boxes (pixels) + sigmoid(conf, cls)
    ostg[tid * NCH + 0] = (s0 + gx) * 8.0f;
    ostg[tid * NCH + 1] = (s1 + gy) * 8.0f;
    ostg[tid * NCH + 2] = __expf(v[2]) * awpx[a];
    ostg[tid * NCH + 3] = __expf(v[3]) * ahpx[a];
#pragma unroll
    for (int c = 0; c < NCLS + 1; ++c) ostg[tid * NCH + 4 + c] = sc[c];

    // ---- loss partials
    const int   cell  = slab * CELLS + pos;
    const int   tw    = twin[cell];
    const float nb    = noobj[cell];
    const float tconf = (tw >= 0) ? 1.0f : 0.0f;
    const float bconf = bce(sc[0], tconf);

    float p_nn = bconf * nb;   // noobj BCE numerator
    float p_nc = nb;           // noobj mask count
    float p_on = 0.0f;         // obj BCE numerator
    float p_bb = 0.0f;         // bbox SE numerator
    float p_cl = 0.0f;         // cls BCE numerator
    float p_oc = 0.0f;         // obj cell count

    if (tw >= 0) {
        p_on = bconf;
        p_oc = 1.0f;
        const float* ti = tinfo + tw * 8;
        float d0 = s0 - ti[0], d1 = s1 - ti[1];
        float d2 = v[2] - ti[2], d3 = v[3] - ti[3];
        p_bb = d0 * d0 + d1 * d1 + d2 * d2 + d3 * d3;
        int lbl = (int)ti[4];
        const float off = 0.05f / 21.0f;              // SMOOTH / NC
#pragma unroll
        for (int c = 0; c < NCLS; ++c) {
            float tc = (c == lbl) ? (0.95f + off) : off;
            p_cl += bce(sc[1 + c], tc);
        }
    }

    __syncthreads();   // all of ostg written

    // ---- output flush: 256*26 = 6656 consecutive floats
    const int64_t base = ((int64_t)slab * CELLS + col0) * NCH;
#if USE_TDM_STORE
    {
        // 1-D tile DMA: LDS staging -> global. The wave retires into S_ENDPGM's
        // implicit wait-idle (covers TENSORcnt); LDS lives until the whole
        // workgroup ends, so the DMA's LDS reads are safe.
        uint64_t ga  = (uint64_t)(void*)out + (uint64_t)base * 4ull;
        uint32_t lds = (uint32_t)(uint64_t)(void*)&ostg[0];
        v4u g0 = tdm_g0(lds, ga);
        v8i g1 = tdm_g1(NCH * CHUNK, 1, NCH * CHUNK);
        v4i gz4 = {0, 0, 0, 0};
        v8i gz8 = {0, 0, 0, 0, 0, 0, 0, 0};
        if (tid == 0)
            __builtin_amdgcn_tensor_store_from_lds(g0, g1, gz4, gz4, gz8, 0);
    }
#else
    for (int i = tid; i < NCH * CHUNK; i += 256) out[base + i] = ostg[i];
#endif

    // ---- deterministic block reduction of 6 partials
    float pv[6] = {p_nn, p_nc, p_on, p_bb, p_cl, p_oc};
    const int lane = tid & 31, wv = tid >> 5;
#pragma unroll
    for (int k = 0; k < 6; ++k) {
        float s = pv[k];
        for (int o = 16; o > 0; o >>= 1) s += __shfl_down(s, o, 32);
        if (lane == 0) red[wv][k] = s;
    }
    __syncthreads();
    if (tid < 6) {
        float s = 0.0f;
#pragma unroll
        for (int w = 0; w < 8; ++w) s += red[w][tid];
        partials[blk * 6 + tid] = s;
    }
}

// ---------------------------------------------------------------- K3: finalize
__global__ __launch_bounds__(256) void k_final(const float* __restrict__ partials,
                                               float* __restrict__ loss_out) {
    __shared__ float red[8][6];
    const int tid = threadIdx.x;
    float acc[6] = {0, 0, 0, 0, 0, 0};
    for (int i = tid; i < NBLK_MAIN; i += 256) {
        const float* p = partials + i * 6;
#pragma unroll
        for (int k = 0; k < 6; ++k) acc[k] += p[k];
    }
    const int lane = tid & 31, wv = tid >> 5;
#pragma unroll
    for (int k = 0; k < 6; ++k) {
        float s = acc[k];
        for (int o = 16; o > 0; o >>= 1) s += __shfl_down(s, o, 32);
        if (lane == 0) red[wv][k] = s;
    }
    __syncthreads();
    if (tid == 0) {
        float tot[6];
#pragma unroll
        for (int k = 0; k < 6; ++k) {
            float s = 0.0f;
            for (int w = 0; w < 8; ++w) s += red[w][k];
            tot[k] = s;
        }
        // tot: 0=noobj_num 1=noobj_cnt 2=obj_num 3=bbox_num 4=cls_num 5=obj_cnt
        float loss = tot[3] / fmaxf(4.0f  * tot[5], 1e-16f)   // bbox
                   + tot[2] / fmaxf(tot[5],         1e-16f)   // obj
                   + tot[0] / fmaxf(tot[1],         1e-16f)   // noobj
                   + tot[4] / fmaxf(21.0f * tot[5], 1e-16f);  // cls
        *loss_out = loss;
    }
}

// ---------------------------------------------------------------- launcher
extern "C" void kernel_launch(void* const* d_in, const int* in_sizes, int n_in,
                              void* d_out, int out_size, void* d_ws, size_t ws_size,
                              hipStream_t stream) {
    (void)in_sizes; (void)n_in; (void)ws_size;
    const float* x  = (const float*)d_in[0];   // (64, 78, 64, 64) f32
    const float* tg = (const float*)d_in[1];   // (2048, 6) f32
    float* out = (float*)d_out;                // 20,447,232 outputs + 1 loss

    // workspace layout (≈ 6.43 MB)
    char*  ws       = (char*)d_ws;
    int*   twin     = (int*)ws;                              // 786432 * 4
    float* noobj    = (float*)(ws + 3145728);                // 786432 * 4
    float* tinfo    = (float*)(ws + 6291456);                // 2048 * 8 * 4
    float* partials = (float*)(ws + 6291456 + 65536);        // 3072 * 6 * 4

    k_init   <<<NCELL / 256, 256, 0, stream>>>(twin, noobj);
    k_targets<<<NTGT / 256, 256, 0, stream>>>(tg, twin, noobj, tinfo);
    k_main   <<<NBLK_MAIN, 256, 0, stream>>>(x, out, twin, noobj, tinfo, partials);
    k_final  <<<1, 256, 0, stream>>>(partials, out + (out_size - 1));
}